// aAsynGCNLayer_70188355551845
// MI455X (gfx1250) — compile-verified
//
#include <hip/hip_runtime.h>

#define GCN_N 50000
#define GCN_E 800000
#define GCN_D 128
#define GCN_MB 5   // 16-row tiles per wave; 50000/16 = 3125 = 625*5

typedef float v2f __attribute__((ext_vector_type(2)));
typedef float v8f __attribute__((ext_vector_type(8)));

// ---------------------------------------------------------------------------
// WMMA f32 GEMM:  Y[N x 128] = X[N x 128] @ W[128 x 128]  (+ optional bias)
// Each wave computes GCN_MB stacked 16x16 output tiles (same 16 columns),
// so each B operand load is reused by GCN_MB V_WMMA_F32_16X16X4_F32 issues.
// Block = 256 threads = 8 waves -> the 8 column tiles of an 80-row stripe.
// A layout (ISA 7.12.2, 32-bit A 16x4): lane = M + 16*(K>>1), vgpr = K&1
// B layout (4x16):                      lane = N + 16*(K>>1), vgpr = K&1
// C/D layout: vgpr v, lanes 0-15 -> M=v, lanes 16-31 -> M=v+8, N = lane&15
// EXEC all-ones: no divergence, exact grid coverage (WMMA requirement).
// ---------------------------------------------------------------------------
__global__ __launch_bounds__(256) void gcn_gemm_wmma_f32(
    const float* __restrict__ X, const float* __restrict__ W,
    const float* __restrict__ bias, float* __restrict__ Y, int has_bias) {
  const int lane = threadIdx.x & 31;
  const int wave = threadIdx.x >> 5;            // 0..7 -> column tile
  const int m0   = blockIdx.x * (16 * GCN_MB);  // 80-row stripe
  const int n0   = wave << 4;
  const int half = lane >> 4;                   // 0 or 1 (selects K pair)
  const int l16  = lane & 15;

  const float* arow = X + (size_t)(m0 + l16) * GCN_D;
  v8f c[GCN_MB];
#pragma unroll
  for (int t = 0; t < GCN_MB; ++t) c[t] = (v8f){};

#pragma unroll
  for (int k = 0; k < GCN_D; k += 4) {
    const int ka = k + 2 * half;
    v2f b;
    b.x = W[(size_t)ka * GCN_D + n0 + l16];
    b.y = W[(size_t)(ka + 1) * GCN_D + n0 + l16];
#pragma unroll
    for (int t = 0; t < GCN_MB; ++t) {
      v2f a;
      a.x = arow[(size_t)t * 16 * GCN_D + ka];
      a.y = arow[(size_t)t * 16 * GCN_D + ka + 1];
      // 8 args: (neg_a, A, neg_b, B, c_mod, C, reuse_a, reuse_b)
      c[t] = __builtin_amdgcn_wmma_f32_16x16x4_f32(false, a, false, b,
                                                   (short)0, c[t], false, false);
    }
  }

#pragma unroll
  for (int t = 0; t < GCN_MB; ++t) {
#pragma unroll
    for (int v = 0; v < 8; ++v) {
      const int row = m0 + t * 16 + v + 8 * half;
      const int col = n0 + l16;
      float val = c[t][v];
      if (has_bias) val += bias[col];
      Y[(size_t)row * GCN_D + col] = val;
    }
  }
}

// deg[i] = 1.0 (self-loop contribution)
__global__ __launch_bounds__(256) void gcn_deg_init(float* __restrict__ deg, int n) {
  int i = blockIdx.x * 256 + threadIdx.x;
  if (i < n) deg[i] = 1.0f;
}

// deg[dst[e]] += 1 over all edges (resolves in L2 atomic units)
__global__ __launch_bounds__(256) void gcn_deg_count(const int* __restrict__ dst,
                                                     float* __restrict__ deg, int e) {
  int i = blockIdx.x * 256 + threadIdx.x;
  if (i < e) atomicAdd(&deg[dst[i]], 1.0f);
}

// in place deg -> rsqrt(deg)   (deg >= 1 always because of self-loops)
__global__ __launch_bounds__(256) void gcn_rsqrt(float* __restrict__ deg, int n) {
  int i = blockIdx.x * 256 + threadIdx.x;
  if (i < n) deg[i] = rsqrtf(deg[i]);
}

// self-loop term: prop[i,:] = h[i,:] * dinv[i]^2  (float4; fully initializes prop)
__global__ __launch_bounds__(256) void gcn_selfloop(const float* __restrict__ H,
                                                    const float* __restrict__ dinv,
                                                    float* __restrict__ prop, int n4) {
  int i = blockIdx.x * 256 + threadIdx.x;
  if (i >= n4) return;
  const int node = i >> 5;  // 32 float4 per 128-wide row
  const float c  = dinv[node] * dinv[node];
  float4 h = ((const float4*)H)[i];
  float4 o;
  o.x = h.x * c; o.y = h.y * c; o.z = h.z * c; o.w = h.w * c;
  ((float4*)prop)[i] = o;
}

// one wave per edge, lane handles 4 consecutive features (128 = 32*4).
// h (25.6MB) and prop (25.6MB) both fit in the 192MB L2 -> random gather and
// atomic scatter stay on-chip.
__global__ __launch_bounds__(256) void gcn_scatter(const float* __restrict__ H,
                                                   const int* __restrict__ src,
                                                   const int* __restrict__ dst,
                                                   const float* __restrict__ dinv,
                                                   float* __restrict__ prop, int e) {
  int g = blockIdx.x * 256 + threadIdx.x;
  int edge = g >> 5;
  int lane = g & 31;
  if (edge >= e) return;
  const int s = src[edge];
  const int d = dst[edge];
  const float coef = dinv[s] * dinv[d];
  float4 h = ((const float4*)(H + (size_t)s * GCN_D))[lane];
  float* pd = prop + (size_t)d * GCN_D + lane * 4;
  atomicAdd(pd + 0, h.x * coef);
  atomicAdd(pd + 1, h.y * coef);
  atomicAdd(pd + 2, h.z * coef);
  atomicAdd(pd + 3, h.w * coef);
}

// out += relu(prop + bias)    (float4 vectorized)
__global__ __launch_bounds__(256) void gcn_accum_relu(const float* __restrict__ prop,
                                                      const float* __restrict__ bias,
                                                      float* __restrict__ out, int n4) {
  int i = blockIdx.x * 256 + threadIdx.x;
  if (i >= n4) return;
  const int c4 = (i & 31) * 4;
  float4 p = ((const float4*)prop)[i];
  float4 b = *(const float4*)(bias + c4);
  float4 o = ((float4*)out)[i];
  o.x += fmaxf(p.x + b.x, 0.0f);
  o.y += fmaxf(p.y + b.y, 0.0f);
  o.z += fmaxf(p.z + b.z, 0.0f);
  o.w += fmaxf(p.w + b.w, 0.0f);
  ((float4*)out)[i] = o;
}

extern "C" void kernel_launch(void* const* d_in, const int* in_sizes, int n_in,
                              void* d_out, int out_size, void* d_ws, size_t ws_size,
                              hipStream_t stream) {
  (void)in_sizes; (void)n_in; (void)out_size; (void)ws_size;
  const float* x0    = (const float*)d_in[0];
  const float* xh[3] = {(const float*)d_in[1], (const float*)d_in[2],
                        (const float*)d_in[3]};
  const int*   ei[3] = {(const int*)d_in[4], (const int*)d_in[5],
                        (const int*)d_in[6]};
  const float* w_ego = (const float*)d_in[7];
  const float* b_ego = (const float*)d_in[8];
  const float* w[3]  = {(const float*)d_in[9], (const float*)d_in[11],
                        (const float*)d_in[13]};
  const float* b[3]  = {(const float*)d_in[10], (const float*)d_in[12],
                        (const float*)d_in[14]};
  float* out = (float*)d_out;

  char* ws = (char*)d_ws;
  float* dinv = (float*)ws;                                  // N floats
  float* hbuf = (float*)(ws + (size_t)(1u << 20));           // N*D floats (25.6MB)
  float* prop = (float*)(ws + (size_t)(1u << 20) + (size_t)(27u << 20));

  const int n_blk    = (GCN_N + 255) / 256;
  const int e_blk    = (GCN_E + 255) / 256;
  const int nd4      = GCN_N * (GCN_D / 4);                  // float4 count
  const int nd4_blk  = (nd4 + 255) / 256;
  const int gemm_blk = GCN_N / (16 * GCN_MB);                // 625, exact
  const int scat_blk = (GCN_E * 32 + 255) / 256;             // wave per edge

  // hidden = x0 @ w_ego + b_ego
  gcn_gemm_wmma_f32<<<gemm_blk, 256, 0, stream>>>(x0, w_ego, b_ego, out, 1);

  for (int hop = 0; hop < 3; ++hop) {
    const int* src = ei[hop];
    const int* dst = ei[hop] + GCN_E;
    gcn_deg_init<<<n_blk, 256, 0, stream>>>(dinv, GCN_N);
    gcn_deg_count<<<e_blk, 256, 0, stream>>>(dst, dinv, GCN_E);
    gcn_rsqrt<<<n_blk, 256, 0, stream>>>(dinv, GCN_N);
    gcn_gemm_wmma_f32<<<gemm_blk, 256, 0, stream>>>(xh[hop], w[hop], nullptr,
                                                    hbuf, 0);
    gcn_selfloop<<<nd4_blk, 256, 0, stream>>>(hbuf, dinv, prop, nd4);
    gcn_scatter<<<scat_blk, 256, 0, stream>>>(hbuf, src, dst, dinv, prop, GCN_E);
    gcn_accum_relu<<<nd4_blk, 256, 0, stream>>>(prop, b[hop], out, nd4);
  }
}